// LSTMAE_88545045774784
// MI455X (gfx1250) — compile-verified
//
#include <hip/hip_runtime.h>
#include <hip/hip_bf16.h>
#include <cstddef>

typedef float v2f __attribute__((ext_vector_type(2)));
typedef float v8f __attribute__((ext_vector_type(8)));

#define B_TOT 256
#define TX 512
#define TS 512
#define EHD 3
#define DHD 128
#define NGATE 512          // 4*DHD
#define HSTR 132           // padded row stride (floats) -> bank-conflict-free frags

__device__ __forceinline__ float sigm_(float x) {
    return 1.0f / (1.0f + __expf(-x));
}
__device__ __forceinline__ float tanh_(float x) {
    // overflow-safe tanh via exp of -2|x|
    float e = __expf(-2.0f * fabsf(x));
    float r = (1.0f - e) / (1.0f + e);
    return copysignf(r, x);
}

__device__ __forceinline__ v8f wmma_f32(v2f a, v2f b, v8f c) {
    // V_WMMA_F32_16X16X4_F32 : D = A(16x4) * B(4x16) + C(16x16), all fp32
    return __builtin_amdgcn_wmma_f32_16x16x4_f32(
        /*neg_a=*/false, a, /*neg_b=*/false, b,
        /*c_mod=*/(short)0, c, /*reuse_a=*/false, /*reuse_b=*/false);
}

// ---------------------------------------------------------------------------
// Encoder: hidden=3 LSTM. 8 blocks x 32 threads -> 1 wave per WGP (min serial
// latency); one thread per batch row, state in registers. Writes z -> d_out[0..255].
// ---------------------------------------------------------------------------
__global__ __launch_bounds__(32) void lstm_encoder_kernel(
    const float* __restrict__ x,      // [B, TX, 8]
    const float* __restrict__ h0,     // [B, 3]
    const float* __restrict__ c0,     // [B, 3]
    const float* __restrict__ Wih,    // [12, 8]
    const float* __restrict__ Whh,    // [12, 3]
    const float* __restrict__ bih,    // [12]
    const float* __restrict__ bhh,    // [12]
    const float* __restrict__ Wfc,    // [1, 3]
    const float* __restrict__ bfc,    // [1]
    float* __restrict__ z_out)        // d_out + 0
{
    __shared__ float s_wih[12 * 8];
    __shared__ float s_whh[12 * 3];
    __shared__ float s_b[12];
    __shared__ float s_wf[3];
    __shared__ float s_bf;

    const int tid = threadIdx.x;
    for (int i = tid; i < 96; i += 32) s_wih[i] = Wih[i];
    for (int i = tid; i < 36; i += 32) s_whh[i] = Whh[i];
    if (tid < 12) s_b[tid]  = bih[tid] + bhh[tid];
    if (tid < 3)  s_wf[tid] = Wfc[tid];
    if (tid == 0) s_bf      = bfc[0];
    __syncthreads();

    const int b = blockIdx.x * 32 + tid;   // one thread per batch row
    float h[3], c[3];
#pragma unroll
    for (int j = 0; j < 3; ++j) { h[j] = h0[b * 3 + j]; c[j] = c0[b * 3 + j]; }

    const float* xb = x + (size_t)b * TX * 8;
    for (int t = 0; t < TX; ++t) {
        float xv[8];
        const float* xr = xb + (size_t)t * 8;
#pragma unroll
        for (int j = 0; j < 8; ++j) xv[j] = xr[j];

        float g[12];
#pragma unroll
        for (int k = 0; k < 12; ++k) {
            float a = s_b[k];
#pragma unroll
            for (int j = 0; j < 8; ++j) a += xv[j] * s_wih[k * 8 + j];
#pragma unroll
            for (int j = 0; j < 3; ++j) a += h[j] * s_whh[k * 3 + j];
            g[k] = a;
        }
#pragma unroll
        for (int j = 0; j < 3; ++j) {
            float ig = sigm_(g[j]);
            float fg = sigm_(g[3 + j]);
            float gg = tanh_(g[6 + j]);
            float og = sigm_(g[9 + j]);
            c[j] = fg * c[j] + ig * gg;
            h[j] = og * tanh_(c[j]);
        }
    }
    z_out[b] = s_wf[0] * h[0] + s_wf[1] * h[1] + s_wf[2] * h[2] + s_bf;
}

// ---------------------------------------------------------------------------
// traj_hat = s @ Wfc_d^T + bfc_d   (pure HBM streaming, ~4 MB total)
// ---------------------------------------------------------------------------
__global__ __launch_bounds__(256) void traj_fc_kernel(
    const float* __restrict__ s,     // [B, TS, 6]
    const float* __restrict__ W,     // [2, 6]
    const float* __restrict__ bias,  // [2]
    float* __restrict__ out)         // d_out + 256, [B, TS, 2]
{
    int id = blockIdx.x * 256 + threadIdx.x;
    if (id >= B_TOT * TS) return;
    const float* sr = s + (size_t)id * 6;
    float sv[6];
#pragma unroll
    for (int j = 0; j < 6; ++j) sv[j] = sr[j];
#pragma unroll
    for (int k = 0; k < 2; ++k) {
        float a = bias[k];
#pragma unroll
        for (int j = 0; j < 6; ++j) a += sv[j] * W[k * 6 + j];
        out[(size_t)id * 2 + k] = a;
    }
}

// ---------------------------------------------------------------------------
// Decoder: hidden=128 LSTM via fp32 WMMA. 16 blocks x 16 batch rows.
// LDS: Whh[512][132] + Wih_pad[512][8] + bias[512] + H[16][132] + xt[2][16][8]
// Wave w (of 8) owns hidden cols 16w..16w+15 for all 4 gates -> c stays in regs.
// Gate-i Whh B-frags live in registers (LDS bandwidth < matrix-pipe bound);
// the f/g/o B-frags + A-frags use an explicit 2-stage software pipeline
// (loads issued 2 K-chunks ahead of use) so LDS latency hides behind WMMAs.
// xt is double-buffered with register prefetch of s(t+1).
// ---------------------------------------------------------------------------
static constexpr int OFF_WHH  = 0;
static constexpr int OFF_WIH  = OFF_WHH + NGATE * HSTR;   // 67584
static constexpr int OFF_BIAS = OFF_WIH + NGATE * 8;      // 71680
static constexpr int OFF_H    = OFF_BIAS + NGATE;         // 72192
static constexpr int OFF_XT   = OFF_H + 16 * HSTR;        // 74304 (2 buffers x 128)
static constexpr int SMEM_FLOATS = OFF_XT + 2 * 16 * 8;   // 74560
static constexpr size_t SMEM_BYTES = (size_t)SMEM_FLOATS * sizeof(float); // 298240 < 320K

__global__ __launch_bounds__(256) void lstm_decoder_kernel(
    const float* __restrict__ s,      // [B, TS, 6]
    const float* __restrict__ z,      // [B]  (d_out z-region, written by encoder)
    const float* __restrict__ Wih,    // [512, 7]
    const float* __restrict__ Whh,    // [512, 128]
    const float* __restrict__ bih,    // [512]
    const float* __restrict__ bhh,    // [512]
    float* __restrict__ hout)         // d_out + 256 + B*TS*2, [B, 128]
{
    extern __shared__ float smem[];
    const int tid = threadIdx.x;
    const int b0  = blockIdx.x * 16;

    // ---- stage weights into LDS (once) ----
    for (int i = tid; i < NGATE * DHD; i += 256)
        smem[OFF_WHH + (i >> 7) * HSTR + (i & 127)] = Whh[i];
    for (int i = tid; i < NGATE * 8; i += 256) {
        int j = i & 7;
        smem[OFF_WIH + i] = (j < 7) ? Wih[(i >> 3) * 7 + j] : 0.0f;
    }
    for (int i = tid; i < NGATE; i += 256)
        smem[OFF_BIAS + i] = bih[i] + bhh[i];
    for (int i = tid; i < 16 * HSTR; i += 256)
        smem[OFF_H + i] = 0.0f;                       // h_0 = 0
    if (tid < 16) {
        float zv = z[b0 + tid];
        // z & K-pad are constant across t: set in both ping-pong buffers
        smem[OFF_XT + tid * 8 + 6] = zv;
        smem[OFF_XT + tid * 8 + 7] = 0.0f;
        smem[OFF_XT + 128 + tid * 8 + 6] = zv;
        smem[OFF_XT + 128 + tid * 8 + 7] = 0.0f;
        // stage s(t=0) into buffer 0
        const float* sr = s + (size_t)(b0 + tid) * TS * 6;
#pragma unroll
        for (int j = 0; j < 6; ++j) smem[OFF_XT + tid * 8 + j] = sr[j];
    }
    __syncthreads();

    const int lane = tid & 31;
    const int wave = tid >> 5;          // 0..7
    const int col  = lane & 15;         // N index (and A-row index m for loads)
    const int hi   = lane >> 4;         // half-wave
    const int koff = hi * 2;            // K sub-offset per ISA f32 A/B layout
    const int hc0  = wave * 16;         // this wave's hidden-column group
    const int n0i = hc0, n0f = DHD + hc0, n0g = 2 * DHD + hc0, n0o = 3 * DHD + hc0;

    // LDS base indices for the streamed fragments (constant per lane)
    const int baseA = OFF_H   + col * HSTR + koff;
    const int baseF = OFF_WHH + (n0f + col) * HSTR + koff;
    const int baseG = OFF_WHH + (n0g + col) * HSTR + koff;
    const int baseO = OFF_WHH + (n0o + col) * HSTR + koff;

    // ---- loop-invariant register preloads ----
    const float bi_c = smem[OFF_BIAS + n0i + col];
    const float bf_c = smem[OFF_BIAS + n0f + col];
    const float bg_c = smem[OFF_BIAS + n0g + col];
    const float bo_c = smem[OFF_BIAS + n0o + col];

    v2f wfi[2], wff[2], wfg[2], wfo[2];           // Wih B-frags (K=8)
#pragma unroll
    for (int c2 = 0; c2 < 2; ++c2) {
        wfi[c2] = *(const v2f*)&smem[OFF_WIH + (n0i + col) * 8 + c2 * 4 + koff];
        wff[c2] = *(const v2f*)&smem[OFF_WIH + (n0f + col) * 8 + c2 * 4 + koff];
        wfg[c2] = *(const v2f*)&smem[OFF_WIH + (n0g + col) * 8 + c2 * 4 + koff];
        wfo[c2] = *(const v2f*)&smem[OFF_WIH + (n0o + col) * 8 + c2 * 4 + koff];
    }
    v2f whi[32];                                   // Whh B-frags, gate i only
#pragma unroll
    for (int kc = 0; kc < 32; ++kc)
        whi[kc] = *(const v2f*)&smem[OFF_WHH + (n0i + col) * HSTR + kc * 4 + koff];

    float cfrag[8];
#pragma unroll
    for (int r = 0; r < 8; ++r) cfrag[r] = 0.0f;   // c_0 = 0

    // per-thread prefetch pointer for s rows (advances by one row per step)
    const float* sptr = (tid < 16) ? (s + ((size_t)(b0 + tid) * TS + 1) * 6) : s;

    float rv[6];
    for (int t = 0; t < TS; ++t) {
        // prefetch s(t+1) into registers (latency hidden by WMMA phase)
        if (tid < 16 && (t + 1) < TS) {
#pragma unroll
            for (int j = 0; j < 6; ++j) rv[j] = sptr[j];
        }

        v8f ai, af, ag, ao;
#pragma unroll
        for (int r = 0; r < 8; ++r) { ai[r] = bi_c; af[r] = bf_c; ag[r] = bg_c; ao[r] = bo_c; }

        // x contribution: K = 8 (7 real + 1 zero pad)
        const float* xbuf = &smem[OFF_XT + (t & 1) * 128];
#pragma unroll
        for (int c2 = 0; c2 < 2; ++c2) {
            v2f a = *(const v2f*)&xbuf[col * 8 + c2 * 4 + koff];
            ai = wmma_f32(a, wfi[c2], ai);
            af = wmma_f32(a, wff[c2], af);
            ag = wmma_f32(a, wfg[c2], ag);
            ao = wmma_f32(a, wfo[c2], ao);
        }

        // recurrent contribution: K = 128 -> 32 chunks x 4 gates, with an
        // explicit 2-chunk-deep software pipeline for the streamed fragments
        v2f aC  = *(const v2f*)&smem[baseA];
        v2f bfC = *(const v2f*)&smem[baseF];
        v2f bgC = *(const v2f*)&smem[baseG];
        v2f boC = *(const v2f*)&smem[baseO];
        v2f aN  = *(const v2f*)&smem[baseA + 4];
        v2f bfN = *(const v2f*)&smem[baseF + 4];
        v2f bgN = *(const v2f*)&smem[baseG + 4];
        v2f boN = *(const v2f*)&smem[baseO + 4];
#pragma unroll
        for (int kc = 0; kc < 32; ++kc) {
            v2f a2, bf2, bg2, bo2;
            if (kc < 30) {
                const int o2 = (kc + 2) * 4;
                a2  = *(const v2f*)&smem[baseA + o2];
                bf2 = *(const v2f*)&smem[baseF + o2];
                bg2 = *(const v2f*)&smem[baseG + o2];
                bo2 = *(const v2f*)&smem[baseO + o2];
            }
            ai = wmma_f32(aC, whi[kc], ai);
            af = wmma_f32(aC, bfC, af);
            ag = wmma_f32(aC, bgC, ag);
            ao = wmma_f32(aC, boC, ao);
            aC = aN;  bfC = bfN;  bgC = bgN;  boC = boN;
            aN = a2;  bfN = bf2;  bgN = bg2;  boN = bo2;
        }
        __syncthreads();   // all reads of H(t-1)/xt(t) done before overwrite

        // nonlinearities + state update; write h(t) back to LDS (D-frag layout)
#pragma unroll
        for (int r = 0; r < 8; ++r) {
            float ig = sigm_(ai[r]);
            float fg = sigm_(af[r]);
            float gg = tanh_(ag[r]);
            float og = sigm_(ao[r]);
            float cr = fg * cfrag[r] + ig * gg;
            cfrag[r] = cr;
            smem[OFF_H + (r + hi * 8) * HSTR + hc0 + col] = og * tanh_(cr);
        }
        // commit prefetched s(t+1) into the other xt buffer
        if (tid < 16 && (t + 1) < TS) {
            float* xb2 = &smem[OFF_XT + ((t + 1) & 1) * 128];
#pragma unroll
            for (int j = 0; j < 6; ++j) xb2[tid * 8 + j] = rv[j];
            sptr += 6;
        }
        __syncthreads();   // H(t) and xt(t+1) visible for next iteration
    }

    // emit final h_dec for this block's 16 rows
    for (int i = tid; i < 16 * DHD; i += 256) {
        int m = i >> 7, n = i & 127;
        hout[(size_t)(b0 + m) * DHD + n] = smem[OFF_H + m * HSTR + n];
    }
}

// ---------------------------------------------------------------------------
extern "C" void kernel_launch(void* const* d_in, const int* in_sizes, int n_in,
                              void* d_out, int out_size, void* d_ws, size_t ws_size,
                              hipStream_t stream) {
    (void)in_sizes; (void)n_in; (void)out_size; (void)d_ws; (void)ws_size;
    const float* x     = (const float*)d_in[0];
    const float* s     = (const float*)d_in[1];
    const float* h0    = (const float*)d_in[2];
    const float* c0    = (const float*)d_in[3];
    const float* We_ih = (const float*)d_in[4];
    const float* We_hh = (const float*)d_in[5];
    const float* be_ih = (const float*)d_in[6];
    const float* be_hh = (const float*)d_in[7];
    const float* Wfc_e = (const float*)d_in[8];
    const float* bfc_e = (const float*)d_in[9];
    const float* Wd_ih = (const float*)d_in[10];
    const float* Wd_hh = (const float*)d_in[11];
    const float* bd_ih = (const float*)d_in[12];
    const float* bd_hh = (const float*)d_in[13];
    const float* Wfc_d = (const float*)d_in[14];
    const float* bfc_d = (const float*)d_in[15];

    float* out      = (float*)d_out;
    float* z_out    = out;                          // [256]
    float* traj_out = out + B_TOT;                  // [256*512*2]
    float* hdec_out = out + B_TOT + B_TOT * TS * 2; // [256*128]

    // allow > default dynamic LDS (~291 KB; WGP supports 320 KB)
    (void)hipFuncSetAttribute((const void*)lstm_decoder_kernel,
                              hipFuncAttributeMaxDynamicSharedMemorySize,
                              (int)SMEM_BYTES);

    lstm_encoder_kernel<<<8, 32, 0, stream>>>(x, h0, c0, We_ih, We_hh, be_ih, be_hh,
                                              Wfc_e, bfc_e, z_out);
    traj_fc_kernel<<<(B_TOT * TS + 255) / 256, 256, 0, stream>>>(s, Wfc_d, bfc_d, traj_out);
    lstm_decoder_kernel<<<16, 256, SMEM_BYTES, stream>>>(s, z_out, Wd_ih, Wd_hh,
                                                         bd_ih, bd_hh, hdec_out);
}